// IndexBoolUnsqueezeSqueezeF32_46806553592123
// MI455X (gfx1250) — compile-verified
//
#include <hip/hip_runtime.h>
#include <stdint.h>

#define BLOCK 256
#define EPT 16                 // elements per thread (contiguous run, for stable order)
#define EPB (BLOCK * EPT)      // 4096 elements (16 KB) per block tile
#define SCAN_THREADS 1024

// ---------------------------------------------------------------------------
// Phase 1: per-block count of positives (coalesced b128 loads)
// ---------------------------------------------------------------------------
__global__ void count_pos(const float* __restrict__ x,
                          unsigned* __restrict__ counts,
                          long long N) {
    __shared__ unsigned red[BLOCK];
    long long base = (long long)blockIdx.x * EPB;
    unsigned c = 0;
#pragma unroll
    for (int k = 0; k < EPT / 4; ++k) {
        long long e = base + (long long)(k * BLOCK + threadIdx.x) * 4;
        if (e + 4 <= N) {
            float4 v = *(const float4*)(x + e);
            c += (v.x > 0.f) + (v.y > 0.f) + (v.z > 0.f) + (v.w > 0.f);
        } else {
            for (int j = 0; j < 4; ++j)
                if (e + j < N) c += (x[e + j] > 0.f);
        }
    }
    red[threadIdx.x] = c;
    __syncthreads();
    for (int off = BLOCK / 2; off > 0; off >>= 1) {
        if (threadIdx.x < (unsigned)off) red[threadIdx.x] += red[threadIdx.x + off];
        __syncthreads();
    }
    if (threadIdx.x == 0) counts[blockIdx.x] = red[0];
}

// ---------------------------------------------------------------------------
// Phase 2: single-block exclusive scan over block counts; emits total count
// ---------------------------------------------------------------------------
__global__ void scan_offsets(const unsigned* __restrict__ counts,
                             unsigned* __restrict__ offsets,
                             unsigned* __restrict__ totalOut,
                             float* __restrict__ out,
                             int nblocks, long long N) {
    __shared__ unsigned sc[SCAN_THREADS];
    int t = threadIdx.x;
    int chunk = (nblocks + SCAN_THREADS - 1) / SCAN_THREADS;
    int lo = t * chunk;
    int hi = lo + chunk; if (hi > nblocks) hi = nblocks;
    unsigned s = 0;
    for (int i = lo; i < hi; ++i) s += counts[i];
    sc[t] = s;
    __syncthreads();
    // inclusive Hillis-Steele scan over thread sums (wave-size agnostic)
    for (int off = 1; off < SCAN_THREADS; off <<= 1) {
        unsigned v = (t >= off) ? sc[t - off] : 0u;
        __syncthreads();
        sc[t] += v;
        __syncthreads();
    }
    unsigned run = sc[t] - s;  // exclusive prefix of this thread's chunk
    for (int i = lo; i < hi; ++i) { offsets[i] = run; run += counts[i]; }
    if (t == SCAN_THREADS - 1) {
        unsigned total = sc[SCAN_THREADS - 1];
        totalOut[0] = total;
        out[N] = (float)total;   // second tuple output: the valid count
    }
}

// ---------------------------------------------------------------------------
// Phase 3a: zero the padded tail out[count .. N)
// ---------------------------------------------------------------------------
__global__ void tail_zero(float* __restrict__ out,
                          const unsigned* __restrict__ totalPtr,
                          long long N) {
    long long total = (long long)(*totalPtr);
    long long i4 = ((long long)blockIdx.x * blockDim.x + threadIdx.x) * 4;
    if (i4 >= N) return;
    if (i4 >= total && i4 + 4 <= N) {
        *(float4*)(out + i4) = make_float4(0.f, 0.f, 0.f, 0.f);
    } else {
        for (int j = 0; j < 4; ++j) {
            long long i = i4 + j;
            if (i < N && i >= total) out[i] = 0.f;
        }
    }
}

// ---------------------------------------------------------------------------
// Phase 3b: stable scatter.
//   1) async DMA global tile -> LDS (GLOBAL_LOAD_ASYNC_TO_LDS_B128, ASYNCcnt)
//   2) per-thread count over its contiguous 16-run + block exclusive scan
//   3) compact positives back into LDS at their in-block rank
//   4) coalesced copy-out via async LDS -> global stores
//      (GLOBAL_STORE_ASYNC_FROM_LDS_B32)
// ---------------------------------------------------------------------------
__global__ void scatter_pos(const float* __restrict__ x,
                            const unsigned* __restrict__ offsets,
                            float* __restrict__ out,
                            long long N) {
    __shared__ __align__(16) float tile[EPB];
    __shared__ unsigned sc[BLOCK];
    long long base = (long long)blockIdx.x * EPB;
    int t = threadIdx.x;

    // ---- 1) async global -> LDS: 1024 x b128 chunks, 4 issued per thread
    uint32_t ldsBase = (uint32_t)(uintptr_t)(&tile[0]);
#pragma unroll
    for (int k = 0; k < EPT / 4; ++k) {
        int chunk = k * BLOCK + t;                    // 16-byte chunk id
        long long e = base + (long long)chunk * 4;    // element index
        uint32_t laddr = ldsBase + (uint32_t)chunk * 16u;
        if (e + 4 <= N) {
            const float* g = x + e;
            asm volatile("global_load_async_to_lds_b128 %0, %1, off"
                         :: "v"(laddr), "v"(g) : "memory");
        } else {
            for (int j = 0; j < 4; ++j)
                tile[chunk * 4 + j] = (e + j < N) ? x[e + j] : 0.f;
        }
    }
    asm volatile("s_wait_asynccnt 0x0" ::: "memory");
    __syncthreads();

    // ---- 2) contiguous run per thread -> stable order; count positives
    int lo = t * EPT;
    float vals[EPT];
    unsigned c = 0;
#pragma unroll
    for (int i = 0; i < EPT; ++i) {
        vals[i] = tile[lo + i];
        c += (vals[i] > 0.f);
    }

    sc[t] = c;
    __syncthreads();   // also guarantees all tile reads are complete
    for (int off = 1; off < BLOCK; off <<= 1) {
        unsigned v = (t >= off) ? sc[t - off] : 0u;
        __syncthreads();
        sc[t] += v;
        __syncthreads();
    }
    unsigned blockCount = sc[BLOCK - 1];   // inclusive total for this block
    unsigned p = sc[t] - c;                // my exclusive in-block rank

    // ---- 3) compact positives into LDS (tile is dead; safe to overwrite)
#pragma unroll
    for (int i = 0; i < EPT; ++i) {
        float v = vals[i];
        if (v > 0.f) tile[p++] = v;
    }
    __syncthreads();

    // ---- 4) coalesced copy-out: LDS -> global via async stores
    unsigned basePos = offsets[blockIdx.x];
    for (unsigned i = t; i < blockCount; i += BLOCK) {
        uint32_t laddr = ldsBase + i * 4u;
        float* g = out + basePos + i;
        asm volatile("global_store_async_from_lds_b32 %0, %1, off"
                     :: "v"(g), "v"(laddr) : "memory");
    }
    asm volatile("s_wait_asynccnt 0x0" ::: "memory");
}

// ---------------------------------------------------------------------------
extern "C" void kernel_launch(void* const* d_in, const int* in_sizes, int n_in,
                              void* d_out, int out_size, void* d_ws, size_t ws_size,
                              hipStream_t stream) {
    const float* x = (const float*)d_in[0];
    long long N = (long long)in_sizes[0];
    float* out = (float*)d_out;

    int nblocks = (int)((N + EPB - 1) / EPB);

    unsigned* ws      = (unsigned*)d_ws;
    unsigned* counts  = ws;                 // [nblocks]
    unsigned* offsets = ws + nblocks;       // [nblocks]
    unsigned* total   = ws + 2 * nblocks;   // [1]

    count_pos<<<nblocks, BLOCK, 0, stream>>>(x, counts, N);
    scan_offsets<<<1, SCAN_THREADS, 0, stream>>>(counts, offsets, total, out, nblocks, N);

    long long tailBlocks = (N + (long long)BLOCK * 4 - 1) / ((long long)BLOCK * 4);
    tail_zero<<<(unsigned)tailBlocks, BLOCK, 0, stream>>>(out, total, N);

    scatter_pos<<<nblocks, BLOCK, 0, stream>>>(x, offsets, out, N);
}